// LookupLanguageModel_64046552318419
// MI455X (gfx1250) — compile-verified
//
#include <hip/hip_runtime.h>
#include <stdint.h>

// n-gram trie constants (match reference exactly)
// V=8192, N=3, C2=32, C3=8
// U = V+1 = 8193, B2 = U*C2 = 262176, B3 = B2*C3 = 2097408
// X = U+B2+1 = 270370, G = B3-1 = 2097407, X+G = 2367777
#define LM_V   8192
#define LM_U   8193
#define LM_XG  2367777

#define SLICES_PER_ROW 4
#define SLICE (LM_V / SLICES_PER_ROW)   // 2048 tokens = 8 KB per block

typedef unsigned int u32x4 __attribute__((ext_vector_type(4)));
typedef int          i32x4 __attribute__((ext_vector_type(4)));
typedef int          i32x8 __attribute__((ext_vector_type(8)));
typedef float        f32x4 __attribute__((ext_vector_type(4)));

__global__ __launch_bounds__(256) void ngram_trie_slice_kernel(
    const int* __restrict__ hist, const int* __restrict__ idxp,
    const int* __restrict__ pointers, const int* __restrict__ ids,
    const float* __restrict__ logs, float* __restrict__ out, int batch)
{
    __shared__ float s_tile[SLICE];  // 8 KB: logs[lo:lo+SLICE] (TDM) -> in-place row build
    __shared__ int   s_c12;          // bigram context node (h0,h1), or -1
    __shared__ float s_base;         // bo12 + bo1
    __shared__ float s_bo12;         // bo12 only

    const int tid  = threadIdx.x;
    const int b    = blockIdx.x / SLICES_PER_ROW;
    const int lo   = (blockIdx.x % SLICES_PER_ROW) * SLICE;
    const int idxv = idxp[0];                       // == N-1 == 2
    const int h0   = hist[(idxv - 2) * batch + b];
    const int h1   = hist[(idxv - 1) * batch + b];

    if (tid == 0) s_c12 = -1;
    __syncthreads();

    if (tid < 32) {
        // ---- TDM: DMA logs[lo : lo+SLICE] (8 KB) into LDS, tracked by TENSORcnt ----
        uint64_t ga      = (uint64_t)(uintptr_t)(logs + lo);
        uint32_t lds_off = (uint32_t)(uintptr_t)&s_tile[0];  // low 32b of LDS-aperture addr = LDS byte offset
        u32x4 g0;
        g0[0] = 1u;                                      // count=1, user D#, no gather
        g0[1] = lds_off;                                 // lds_addr (bytes)
        g0[2] = (uint32_t)ga;                            // global_addr[31:0]
        g0[3] = (uint32_t)((ga >> 32) & 0x01FFFFFFu)     // global_addr[56:32]
              | (2u << 30);                              // type = 2 ("image")
        i32x8 g1;
        g1[0] = (int)(2u << 16);                         // data_size = 4 bytes; no multicast
        g1[1] = (int)((uint32_t)SLICE << 16);            // tensor_dim0[15:0] = 2048
        g1[2] = (int)(1u << 16);                         // tensor_dim0[31:16]=0 | tensor_dim1 = 1
        g1[3] = (int)((uint32_t)SLICE << 16);            // tile_dim0 = 2048
        g1[4] = 0;                                       // tile_dim1/2 unused
        g1[5] = SLICE;                                   // tensor_dim0_stride[31:0]
        g1[6] = 0;
        g1[7] = 0;
        i32x4 g2 = {0, 0, 0, 0};
        i32x4 g3 = {0, 0, 0, 0};
#if __clang_major__ >= 23
        i32x8 g4 = {0, 0, 0, 0, 0, 0, 0, 0};
        __builtin_amdgcn_tensor_load_to_lds(g0, g1, g2, g3, g4, 0);
#else
        __builtin_amdgcn_tensor_load_to_lds(g0, g1, g2, g3, 0);
#endif

        // ---- bigram-context search: child of unigram h0 with token id h1 ----
        int offs  = pointers[h0];
        int nc    = pointers[h0 + 1] - offs + 1;         // == 32
        int child = h0 + offs + tid;
        if (tid < nc && ids[child - LM_U] == h1) s_c12 = child;  // ids distinct per parent

        __builtin_amdgcn_s_wait_tensorcnt(0);            // LDS tile ready for this wave
    }
    __syncthreads();

    if (tid == 0) {
        int   c12  = s_c12;
        float bo12 = (c12 >= 0) ? logs[LM_XG + c12] : 0.0f;  // backoff of (h0,h1) context
        float bo1  = logs[LM_XG + h1];                        // backoff of unigram h1
        s_bo12 = bo12;
        s_base = bo12 + bo1;
    }
    __syncthreads();

    // ---- default path for every v in slice: bo12 + bo1 + logs[v] (in-place in LDS) ----
    {
        const float base = s_base;
        f32x4* t4 = (f32x4*)s_tile;
        #pragma unroll
        for (int i = 0; i < SLICE / 4 / 256; ++i) {
            int j = tid + i * 256;
            f32x4 l = t4[j];
            t4[j] = l + base;                            // vector splat add
        }
    }
    __syncthreads();

    // ---- bigram hits: children of unigram h1 override the default ----
    if (tid < 32) {
        int offs = pointers[h1];
        int nc   = pointers[h1 + 1] - offs + 1;          // == 32
        if (tid < nc) {
            int child = h1 + offs + tid;
            int tok   = ids[child - LM_U] - lo;          // position within this slice
            if ((unsigned)tok < (unsigned)SLICE)
                s_tile[tok] = s_bo12 + logs[child];
        }
    }
    __syncthreads();

    // ---- trigram hits: children of c12 override everything ----
    {
        int c12 = s_c12;
        if (c12 >= 0 && tid < 32) {
            int offs = pointers[c12];
            int nc   = pointers[c12 + 1] - offs + 1;     // == 8
            if (tid < nc) {
                int child = c12 + offs + tid;
                int tok   = ids[child - LM_U] - lo;
                if ((unsigned)tok < (unsigned)SLICE)
                    s_tile[tok] = logs[child];
            }
        }
    }
    __syncthreads();

    // ---- coalesced non-temporal writeout of the finished slice ----
    {
        const f32x4* t4 = (const f32x4*)s_tile;
        f32x4* o4 = (f32x4*)(out + (size_t)b * LM_V + lo);
        #pragma unroll
        for (int i = 0; i < SLICE / 4 / 256; ++i) {
            int j = tid + i * 256;
            f32x4 v = t4[j];
            __builtin_nontemporal_store(v, &o4[j]);
        }
    }
}

extern "C" void kernel_launch(void* const* d_in, const int* in_sizes, int n_in,
                              void* d_out, int out_size, void* d_ws, size_t ws_size,
                              hipStream_t stream) {
    const int*   hist     = (const int*)d_in[0];
    const int*   idxp     = (const int*)d_in[1];
    const int*   pointers = (const int*)d_in[2];
    const int*   ids      = (const int*)d_in[3];
    const float* logs     = (const float*)d_in[4];
    float*       out      = (float*)d_out;
    const int batch = in_sizes[0] / 2;   // hist is (N-1, BATCH) = (2, 128)
    ngram_trie_slice_kernel<<<batch * SLICES_PER_ROW, 256, 0, stream>>>(
        hist, idxp, pointers, ids, logs, out, batch);
}